// CombinedN2NWaveletLoss_31301721653342
// MI455X (gfx1250) — compile-verified
//
#include <hip/hip_runtime.h>
#include <stdint.h>

// =====================================================================
// Fused Noise2Noise + 3-level Haar wavelet sparsity loss for MI455X.
//
//  - One pass over the 33.5 MB input (bandwidth bound: ~1.4 us at 23.3 TB/s).
//  - Checkerboard phases p0/p3 staged to LDS via gfx1250 async-to-LDS loads.
//  - g1 (bilinear upsample of p0) materialized per-tile in LDS, conv3x3 +
//    clip + both MSE terms + full Haar pyramid computed in-registers/LDS.
//  - Per-wave reduction of the 5 partial sums via v_wmma_f32_16x16x4_f32
//    (ones-selector matmul reduction) + 16-lane butterfly, then fp32
//    global atomics into d_ws; a 1-thread kernel applies normalizations.
// =====================================================================

typedef float v2f __attribute__((ext_vector_type(2)));
typedef float v8f __attribute__((ext_vector_type(8)));

#define NIMG 32
#define HW   512
#define TPB  256

// LDS tile geometry (out-tile = 64x64, p0/p3 halo'd 34x34, g1 halo'd 66x66)
#define P0R 34
#define P0P 36   // padded row stride (floats)
#define G1R 66
#define G1P 68   // padded row stride (floats)

#if defined(__gfx1250__) && __has_builtin(__builtin_amdgcn_wmma_f32_16x16x4_f32)
#define HAVE_WMMA_F32X4 1
#else
#define HAVE_WMMA_F32X4 0
#endif

// Bilinear sample of a 256x256 half-res plane held in a halo'd LDS tile.
// Tile origin is (IY-1, IX-1) in half-res coords; row stride P0P.
// jax.image.resize('linear', 256->512): src = y/2 - 0.25, edge-renormalized
// (== index-clamped here since only one tap can fall outside).
__device__ __forceinline__ float bilin_tile(const float* __restrict__ t,
                                            int Y, int X, int IY, int IX) {
  int iy = Y >> 1, ix = X >> 1;
  int r0, r1, c0, c1;
  float wy0, wy1, wx0, wx1;
  if (Y & 1) { r0 = iy;     r1 = iy + 1; wy0 = 0.75f; wy1 = 0.25f; }
  else       { r0 = iy - 1; r1 = iy;     wy0 = 0.25f; wy1 = 0.75f; }
  if (X & 1) { c0 = ix;     c1 = ix + 1; wx0 = 0.75f; wx1 = 0.25f; }
  else       { c0 = ix - 1; c1 = ix;     wx0 = 0.25f; wx1 = 0.75f; }
  r0 = r0 < 0 ? 0 : (r0 > 255 ? 255 : r0);
  r1 = r1 < 0 ? 0 : (r1 > 255 ? 255 : r1);
  c0 = c0 < 0 ? 0 : (c0 > 255 ? 255 : c0);
  c1 = c1 < 0 ? 0 : (c1 > 255 ? 255 : c1);
  r0 -= IY - 1; r1 -= IY - 1; c0 -= IX - 1; c1 -= IX - 1;
  float a = t[r0 * P0P + c0], b = t[r0 * P0P + c1];
  float c = t[r1 * P0P + c0], d = t[r1 * P0P + c1];
  return wy0 * (wx0 * a + wx1 * b) + wy1 * (wx0 * c + wx1 * d);
}

__global__ void init_ws_kernel(float* __restrict__ ws) {
  if (threadIdx.x < 8) ws[threadIdx.x] = 0.0f;
}

__global__ __launch_bounds__(TPB) void n2n_wavelet_fused(
    const float* __restrict__ noisy, const float* __restrict__ wgt,
    float* __restrict__ ws) {
  __shared__ float p0s[P0R * P0P];   // checkerboard phase (even,even)
  __shared__ float p3s[P0R * P0P];   // checkerboard phase (odd,odd)
  __shared__ float g1s[G1R * G1P];   // upsampled p0, out-tile + conv halo
  __shared__ float ll1s[32 * 33];    // Haar level-1 LL
  __shared__ float ll2s[16 * 17];    // Haar level-2 LL

  const int tid  = threadIdx.x;
  const int img  = blockIdx.x >> 6;          // 64 tiles per image
  const int tile = blockIdx.x & 63;
  const int ty = tile >> 3, tx = tile & 7;
  const int OY = ty * 64, OX = tx * 64;      // out-tile origin (full res)
  const int IY = OY >> 1, IX = OX >> 1;      // half-res origin

  const unsigned long long base =
      (unsigned long long)(noisy + (size_t)img * HW * HW);

  // ---- stage p0/p3 (34x34, edge-clamped) via async global->LDS loads ----
  for (int idx = tid; idx < P0R * P0R; idx += TPB) {
    int r = idx / P0R, c = idx % P0R;
    int gr = IY - 1 + r; gr = gr < 0 ? 0 : (gr > 255 ? 255 : gr);
    int gc = IX - 1 + c; gc = gc < 0 ? 0 : (gc > 255 ? 255 : gc);
    unsigned off0 = ((unsigned)(2 * gr) * (unsigned)HW + (unsigned)(2 * gc)) * 4u;
    unsigned off3 = ((unsigned)(2 * gr + 1) * (unsigned)HW + (unsigned)(2 * gc + 1)) * 4u;
    unsigned lds0 = (unsigned)(uintptr_t)&p0s[r * P0P + c];
    unsigned lds3 = (unsigned)(uintptr_t)&p3s[r * P0P + c];
#if defined(__gfx1250__)
    asm volatile("global_load_async_to_lds_b32 %0, %1, %2"
                 :: "v"(lds0), "v"(off0), "s"(base) : "memory");
    asm volatile("global_load_async_to_lds_b32 %0, %1, %2"
                 :: "v"(lds3), "v"(off3), "s"(base) : "memory");
#else
    p0s[r * P0P + c] = noisy[(size_t)img * HW * HW + (2 * gr) * HW + 2 * gc];
    p3s[r * P0P + c] = noisy[(size_t)img * HW * HW + (2 * gr + 1) * HW + 2 * gc + 1];
#endif
  }
#if defined(__gfx1250__)
  asm volatile("s_wait_asynccnt 0x0" ::: "memory");
#endif
  __syncthreads();

  // conv weights (uniform scalar loads)
  float w[9];
#pragma unroll
  for (int i = 0; i < 9; ++i) w[i] = wgt[i];

  // ---- materialize g1 tile (66x66: out tile + 1-px conv halo, zero pad) ----
  for (int idx = tid; idx < G1R * G1R; idx += TPB) {
    int ry = idx / G1R, rx = idx % G1R;
    int Y = OY - 1 + ry, X = OX - 1 + rx;
    float v = 0.0f;                              // conv SAME zero padding
    if (Y >= 0 && Y < HW && X >= 0 && X < HW)
      v = bilin_tile(p0s, Y, X, IY, IX);
    g1s[ry * G1P + rx] = v;
  }
  __syncthreads();

  // ---- main quad loop: conv + clip + n2n terms + Haar level 1 ----
  const float T3 = 50.0f / 255.0f;   // level_idx 1 threshold (coarsest)
  const float T2 = T3 * 0.5f;        // level_idx 2
  const float T1 = T3 * 0.25f;       // level_idx 3 (finest Haar pass)
  float rec = 0.0f, reg = 0.0f, s1 = 0.0f, s2 = 0.0f, s3 = 0.0f;

#pragma unroll
  for (int k = 0; k < 4; ++k) {
    int q = tid * 4 + k;             // 0..1023 quads in 32x32 grid
    int qy = q >> 5, qx = q & 31;
    float o[2][2];
#pragma unroll
    for (int py = 0; py < 2; ++py) {
#pragma unroll
      for (int px = 0; px < 2; ++px) {
        int ly = 2 * qy + py, lx = 2 * qx + px;      // 0..63 within tile
        const float* gp = &g1s[ly * G1P + lx];       // 3x3 top-left
        float conv = w[0] * gp[0]         + w[1] * gp[1]         + w[2] * gp[2]
                   + w[3] * gp[G1P]       + w[4] * gp[G1P + 1]   + w[5] * gp[G1P + 2]
                   + w[6] * gp[2 * G1P]   + w[7] * gp[2 * G1P + 1] + w[8] * gp[2 * G1P + 2];
        float ov = conv < 0.0f ? 0.0f : (conv > 1.0f ? 1.0f : conv);
        float g1c = gp[G1P + 1];
        float g2v = bilin_tile(p3s, OY + ly, OX + lx, IY, IX);
        float dr = ov - g2v; rec += dr * dr;
        float dg = ov - g1c; reg += dg * dg;
        o[py][px] = ov;
      }
    }
    float a = o[0][0], b = o[0][1], c = o[1][0], d = o[1][1];
    float lh = (a - b + c - d) * 0.5f;
    float hl = (a + b - c - d) * 0.5f;
    float hh = (a - b - c + d) * 0.5f;
    s1 += fminf(fabsf(lh), T1) + fminf(fabsf(hl), T1) + fminf(fabsf(hh), T1);
    ll1s[qy * 33 + qx] = (a + b + c + d) * 0.5f;
  }
  __syncthreads();

  // ---- Haar level 2: 16x16 quads, one per thread ----
  {
    int qy = tid >> 4, qx = tid & 15;
    float a = ll1s[(2 * qy) * 33 + 2 * qx];
    float b = ll1s[(2 * qy) * 33 + 2 * qx + 1];
    float c = ll1s[(2 * qy + 1) * 33 + 2 * qx];
    float d = ll1s[(2 * qy + 1) * 33 + 2 * qx + 1];
    float lh = (a - b + c - d) * 0.5f;
    float hl = (a + b - c - d) * 0.5f;
    float hh = (a - b - c + d) * 0.5f;
    s2 += fminf(fabsf(lh), T2) + fminf(fabsf(hl), T2) + fminf(fabsf(hh), T2);
    ll2s[qy * 17 + qx] = (a + b + c + d) * 0.5f;
  }
  __syncthreads();

  // ---- Haar level 3: 8x8 quads, threads 0..63 ----
  if (tid < 64) {
    int qy = tid >> 3, qx = tid & 7;
    float a = ll2s[(2 * qy) * 17 + 2 * qx];
    float b = ll2s[(2 * qy) * 17 + 2 * qx + 1];
    float c = ll2s[(2 * qy + 1) * 17 + 2 * qx];
    float d = ll2s[(2 * qy + 1) * 17 + 2 * qx + 1];
    float lh = (a - b + c - d) * 0.5f;
    float hl = (a + b - c - d) * 0.5f;
    float hh = (a - b - c + d) * 0.5f;
    s3 += fminf(fabsf(lh), T3) + fminf(fabsf(hl), T3) + fminf(fabsf(hh), T3);
  }

  // ---- per-wave reduction of (rec, reg, s1, s2, s3) ----
  const int lane = tid & 31;
#if HAVE_WMMA_F32X4
  // Ones-selector WMMA reduction: A[m,0..1]=1 for m<8, A[m,2..3]=1 for m>=8.
  // With B vgpr0 = q0, vgpr1 = q1:  D[0,n] = q0[n]+q1[n] (lanes 0-15),
  //                                 D[8,n] = q0[n+16]+q1[n+16] (lanes 16-31).
  float sel = ((lane < 8) || (lane >= 24)) ? 1.0f : 0.0f;
  v2f A; A[0] = sel; A[1] = sel;
  v8f Z = {0.f, 0.f, 0.f, 0.f, 0.f, 0.f, 0.f, 0.f};
  v2f B1; B1[0] = rec; B1[1] = reg;
  v2f B2; B2[0] = s1;  B2[1] = s2;
  v2f B3; B3[0] = s3;  B3[1] = 0.0f;
  v8f D1 = __builtin_amdgcn_wmma_f32_16x16x4_f32(false, A, false, B1, (short)0, Z, false, false);
  v8f D2 = __builtin_amdgcn_wmma_f32_16x16x4_f32(false, A, false, B2, (short)0, Z, false, false);
  v8f D3 = __builtin_amdgcn_wmma_f32_16x16x4_f32(false, A, false, B3, (short)0, Z, false, false);
  float r1 = D1[0], r2 = D2[0], r3 = D3[0];
#pragma unroll
  for (int m = 8; m >= 1; m >>= 1) {   // butterfly within each 16-lane half
    r1 += __shfl_xor(r1, m, 32);
    r2 += __shfl_xor(r2, m, 32);
    r3 += __shfl_xor(r3, m, 32);
  }
  if (lane == 0) {
    atomicAdd(&ws[0], r1);  // rec+reg over lanes 0-15
    atomicAdd(&ws[2], r2);  // s1+s2 over lanes 0-15
    atomicAdd(&ws[4], r3);  // s3 over lanes 0-15 only
  } else if (lane == 16) {
    atomicAdd(&ws[1], r1);  // rec+reg over lanes 16-31
    atomicAdd(&ws[3], r2);  // s1+s2 over lanes 16-31
  }
#else
  // fallback: plain 5-step wave butterfly
#pragma unroll
  for (int m = 16; m >= 1; m >>= 1) {
    rec += __shfl_xor(rec, m, 32);
    reg += __shfl_xor(reg, m, 32);
    s1  += __shfl_xor(s1,  m, 32);
    s2  += __shfl_xor(s2,  m, 32);
    s3  += __shfl_xor(s3,  m, 32);
  }
  if (lane == 0) {
    atomicAdd(&ws[0], rec); atomicAdd(&ws[1], reg);
    atomicAdd(&ws[2], s1);  atomicAdd(&ws[3], s2);  atomicAdd(&ws[4], s3);
  }
#endif
}

__global__ void finalize_loss(const float* __restrict__ ws, float* __restrict__ out) {
  const float N   = 32.0f * 512.0f * 512.0f;   // 8388608
  const float N1  = 32.0f * 256.0f * 256.0f;   // level 1 coeff count
  const float N2  = 32.0f * 128.0f * 128.0f;
  const float N3  = 32.0f * 64.0f * 64.0f;
  float n2n = ws[0] / N + 2.0f * (ws[1] / N);           // rec + GAMMA*reg
  float wav = (1.0f / 3.0f) * (ws[2] / (3.0f * N1))     // w=1/3 (level_idx 3)
            + 0.5f          * (ws[3] / (3.0f * N2))     // w=1/2
            +                 (ws[4] / (3.0f * N3));    // w=1
  out[0] = n2n + 0.05f * wav;
}

extern "C" void kernel_launch(void* const* d_in, const int* in_sizes, int n_in,
                              void* d_out, int out_size, void* d_ws, size_t ws_size,
                              hipStream_t stream) {
  const float* noisy = (const float*)d_in[0];   // (32,1,512,512) fp32
  const float* wgt   = (const float*)d_in[1];   // (1,1,3,3) fp32
  float* out = (float*)d_out;                   // 1 fp32 scalar
  float* ws  = (float*)d_ws;

  init_ws_kernel<<<dim3(1), dim3(32), 0, stream>>>(ws);
  n2n_wavelet_fused<<<dim3(NIMG * 64), dim3(TPB), 0, stream>>>(noisy, wgt, ws);
  finalize_loss<<<dim3(1), dim3(1), 0, stream>>>(ws, out);
}